// BaselineDecoder4_18897856103019
// MI455X (gfx1250) — compile-verified
//
#include <hip/hip_runtime.h>
#include <hip/hip_bf16.h>
#include <math.h>

// Problem constants (from reference)
#define T_  64
#define B_  32
#define H_  512
#define S_  64
#define V_  32000
#define G4H (4 * H_)
#define TB  (T_ * B_)

typedef __attribute__((ext_vector_type(2))) float v2f;
typedef __attribute__((ext_vector_type(8))) float v8f;

// LDS B-tile stride: 512 + 4 pad -> lane n hits bank (4n+k)%64, conflict-free.
// 516*4 = 2064 bytes, a multiple of 16 -> every (n,k%4==0) chunk is B128-aligned.
#define LDSB_STRIDE 516

// ---------------------------------------------------------------------------
// Async staging of a 16x512 B-tile (stored transposed [n][k]) into padded LDS
// via GLOBAL_LOAD_ASYNC_TO_LDS_B128 (gfx1250 async memory->LDS path, ASYNCcnt).
// The LDS destination is passed as a *pointer-derived* value so the shared
// array is captured: the memory-clobbered asm is then a legal writer of ldsB
// and the compiler must keep the subsequent ds_load reads (round-2 bug: an
// unrelated integer offset left ldsB "never stored" -> B loads folded to undef).
// Each lane moves one 16-byte chunk per iteration; 2048 chunks / 256 threads.
// Bsrc element (k,n) lives at Bsrc[n*ldb + k].
// ---------------------------------------------------------------------------
__device__ __forceinline__ void load_B_lds_async(float* ldsB,
                                                 const float* __restrict__ Bsrc,
                                                 int ldb) {
  for (int c = threadIdx.x; c < 16 * 128; c += 256) {
    int n = c >> 7;              // row (0..15)
    int k = (c & 127) << 2;      // chunk of 4 floats within the 512-wide row
    float* dst = ldsB + n * LDSB_STRIDE + k;
    // Low 32 bits of a generic LDS pointer == LDS byte offset (ISA 10.2).
    unsigned lds_addr = (unsigned)(size_t)dst;
    const float* g = Bsrc + (size_t)n * ldb + k;
    asm volatile("global_load_async_to_lds_b128 %0, %1, off"
                 :: "v"(lds_addr), "v"(g)
                 : "memory");
  }
  asm volatile("s_wait_asynccnt 0x0" ::: "memory");
}

// ---------------------------------------------------------------------------
// Dual 16x16 f32 WMMA tiles sharing one B tile: each LDS B-fragment feeds two
// v_wmma_f32_16x16x4_f32 ops (halves DS traffic per matrix op).
// A element (m,k) at A*[m*lda + k]; B (k,n) at ldsB[n*LDSB_STRIDE + k].
// Lanes 0-15 hold K={k,k+1}, lanes 16-31 hold K={k+2,k+3} (ISA 16x4 layout).
// ---------------------------------------------------------------------------
__device__ __forceinline__ void wmma_tile2_ldsB(const float* __restrict__ A0,
                                                const float* __restrict__ A1,
                                                int lda,
                                                const float* __restrict__ ldsB,
                                                v8f& acc0, v8f& acc1) {
  const int lane = threadIdx.x & 31;
  const int half = lane >> 4;
  const int l15  = lane & 15;
  const float* arow0 = A0 + l15 * lda;
  const float* arow1 = A1 + l15 * lda;
  const float* brow  = ldsB + l15 * LDSB_STRIDE;
  for (int k = 0; k < H_; k += 4) {
    v2f b  = *(const v2f*)(brow + k + 2 * half);
    v2f a0 = *(const v2f*)(arow0 + k + 2 * half);
    v2f a1 = *(const v2f*)(arow1 + k + 2 * half);
    acc0 = __builtin_amdgcn_wmma_f32_16x16x4_f32(false, a0, false, b, (short)0, acc0,
                                                 false, false);
    acc1 = __builtin_amdgcn_wmma_f32_16x16x4_f32(false, a1, false, b, (short)0, acc1,
                                                 false, false);
  }
}

__device__ __forceinline__ void store_tile(float* __restrict__ Crow, int ldc, v8f acc) {
  const int lane = threadIdx.x & 31;
  const int half = lane >> 4;
  const int l15  = lane & 15;
#pragma unroll
  for (int r = 0; r < 8; ++r) {
    Crow[(r + 8 * half) * ldc + l15] = acc[r];
  }
}

// ---------------------------------------------------------------------------
// Embedding gather: x[t,b,:] = emb[tgt[t,b],:]
// ---------------------------------------------------------------------------
__global__ void __launch_bounds__(256) k_embed(const int* __restrict__ tgt,
                                               const float* __restrict__ emb,
                                               float* __restrict__ x) {
  int idx = blockIdx.x * 256 + threadIdx.x;   // over TB*H
  int h   = idx & (H_ - 1);
  int tb  = idx >> 9;
  x[idx] = emb[(size_t)tgt[tb] * H_ + h];
}

// ---------------------------------------------------------------------------
// Gate pre-projection: G[TB,4H] = X[TB,H] @ W_ih^T + (b_ih + b_hh)
// grid = (4H/16, TB/256); 8 waves x 2 M-tiles each against one LDS B tile.
// ---------------------------------------------------------------------------
__global__ void __launch_bounds__(256)
k_gemm_gates(const float* __restrict__ A, const float* __restrict__ Wih,
             const float* __restrict__ b0, const float* __restrict__ b1,
             float* __restrict__ C) {
  __shared__ __align__(16) float ldsB[16 * LDSB_STRIDE];
  const int nBase = blockIdx.x * 16;
  load_B_lds_async(ldsB, Wih + (size_t)nBase * H_, H_);
  __syncthreads();
  const int wid = threadIdx.x >> 5;
  const int mT0 = blockIdx.y * 16 + wid * 2;   // first of this wave's two M-tiles
  const int l15 = threadIdx.x & 15;
  const float bv = b0[nBase + l15] + b1[nBase + l15];
  v8f acc0, acc1;
#pragma unroll
  for (int r = 0; r < 8; ++r) { acc0[r] = bv; acc1[r] = bv; }
  wmma_tile2_ldsB(A + (size_t)mT0 * 16 * H_, A + (size_t)(mT0 + 1) * 16 * H_, H_,
                  ldsB, acc0, acc1);
  store_tile(C + (size_t)mT0 * 16 * G4H + nBase, G4H, acc0);
  store_tile(C + (size_t)(mT0 + 1) * 16 * G4H + nBase, G4H, acc1);
}

// ---------------------------------------------------------------------------
// One LSTM timestep. grid = 32 blocks, each owns 16 h-columns.
// Waves (8): gate g = wid>>1, mTile = wid&1 (B=32 -> 2 row tiles).
// gates = G[t] + h_in @ W_hh^T ; then pointwise update of c, h.
// ---------------------------------------------------------------------------
__global__ void __launch_bounds__(256)
k_lstm_step(const float* __restrict__ Gt, const float* __restrict__ Whh,
            const float* __restrict__ h_in, float* __restrict__ h_out,
            float* __restrict__ c, float* __restrict__ y_t) {
  __shared__ float lg[4][32][17];
  const int wid  = threadIdx.x >> 5;
  const int g    = wid >> 1;
  const int mT   = wid & 1;
  const int lane = threadIdx.x & 31;
  const int half = lane >> 4;
  const int l15  = lane & 15;
  const int n    = g * H_ + blockIdx.x * 16 + l15;   // column in 4H gate space

  v8f acc;
#pragma unroll
  for (int r = 0; r < 8; ++r)
    acc[r] = Gt[(size_t)(mT * 16 + r + 8 * half) * G4H + n];

  {
    const float* arow = h_in + (size_t)(mT * 16 + l15) * H_;
    const float* brow = Whh + (size_t)n * H_;
    for (int k = 0; k < H_; k += 4) {
      v2f a = *(const v2f*)(arow + k + 2 * half);
      v2f b = *(const v2f*)(brow + k + 2 * half);
      acc = __builtin_amdgcn_wmma_f32_16x16x4_f32(false, a, false, b, (short)0, acc,
                                                  false, false);
    }
  }
#pragma unroll
  for (int r = 0; r < 8; ++r) lg[g][mT * 16 + r + 8 * half][l15] = acc[r];
  __syncthreads();

  for (int idx = threadIdx.x; idx < B_ * 16; idx += 256) {
    int m   = idx >> 4;
    int col = idx & 15;
    int hc  = blockIdx.x * 16 + col;
    float iv = lg[0][m][col];
    float fv = lg[1][m][col];
    float gv = lg[2][m][col];
    float ov = lg[3][m][col];
    float cp = c[m * H_ + hc];
    float ig = 1.0f / (1.0f + __expf(-iv));
    float fg = 1.0f / (1.0f + __expf(-fv));
    float og = 1.0f / (1.0f + __expf(-ov));
    float cn = fg * cp + ig * tanhf(gv);
    float hn = og * tanhf(cn);
    c[m * H_ + hc]     = cn;
    h_out[m * H_ + hc] = hn;
    y_t[m * H_ + hc]   = hn;
  }
}

// ---------------------------------------------------------------------------
// Degenerate attention: softmax weights are exactly uniform over unmasked s
// (exp(NEG - att) == 0 in f32), so ctx[b,h] = mean of enc over {s: mask==0},
// or mean over all s if every s is masked. W1 drops out entirely.
// ---------------------------------------------------------------------------
__global__ void __launch_bounds__(256) k_ctx(const float* __restrict__ enc,
                                             const int* __restrict__ smask,
                                             float* __restrict__ ctx) {
  int idx = blockIdx.x * 256 + threadIdx.x;  // over B*H
  int h = idx & (H_ - 1);
  int b = idx >> 9;
  float su = 0.0f, sa = 0.0f;
  int cnt = 0;
  for (int s = 0; s < S_; ++s) {
    float v = enc[((size_t)s * B_ + b) * H_ + h];
    sa += v;
    if (smask[s * B_ + b] == 0) { su += v; ++cnt; }
  }
  ctx[idx] = (cnt > 0) ? (su / (float)cnt) : (sa / (float)S_);
}

// ---------------------------------------------------------------------------
// ctxW2[B,V] = ctx[B,H] @ W2[:, H:2H]^T  (t-independent half of the vocab proj)
// grid = (V/16, 1); wave 0 computes both M-tiles (M=32), all waves fill LDS.
// ---------------------------------------------------------------------------
__global__ void __launch_bounds__(256)
k_gemm_ctx(const float* __restrict__ ctx, const float* __restrict__ W2,
           float* __restrict__ ctxW2) {
  __shared__ __align__(16) float ldsB[16 * LDSB_STRIDE];
  const int nBase = blockIdx.x * 16;
  load_B_lds_async(ldsB, W2 + (size_t)nBase * (2 * H_) + H_, 2 * H_);
  __syncthreads();
  const int wid = threadIdx.x >> 5;
  if (wid == 0) {
    v8f acc0, acc1;
#pragma unroll
    for (int r = 0; r < 8; ++r) { acc0[r] = 0.0f; acc1[r] = 0.0f; }
    wmma_tile2_ldsB(ctx, ctx + 16 * H_, H_, ldsB, acc0, acc1);
    store_tile(ctxW2 + nBase, V_, acc0);
    store_tile(ctxW2 + (size_t)16 * V_ + nBase, V_, acc1);
  }
}

// ---------------------------------------------------------------------------
// Final vocab projection: out[m,v] = dec[m,:] @ W2[v,:H] + ctxW2[m%32, v]
// M = TB = 2048, N = V = 32000, K = 512. grid = (V/16, TB/256).
// ---------------------------------------------------------------------------
__global__ void __launch_bounds__(256)
k_gemm_out(const float* __restrict__ dec, const float* __restrict__ W2,
           const float* __restrict__ ctxW2, float* __restrict__ out) {
  __shared__ __align__(16) float ldsB[16 * LDSB_STRIDE];
  const int nBase = blockIdx.x * 16;
  load_B_lds_async(ldsB, W2 + (size_t)nBase * (2 * H_), 2 * H_);
  __syncthreads();
  const int wid  = threadIdx.x >> 5;
  const int mT0  = blockIdx.y * 16 + wid * 2;
  const int lane = threadIdx.x & 31;
  const int half = lane >> 4;
  const int l15  = lane & 15;
  v8f acc0, acc1;
#pragma unroll
  for (int r = 0; r < 8; ++r) {
    int m0 = mT0 * 16 + r + 8 * half;          // m = t*32 + b  ->  b = m & 31
    int m1 = (mT0 + 1) * 16 + r + 8 * half;
    acc0[r] = ctxW2[(size_t)(m0 & (B_ - 1)) * V_ + nBase + l15];
    acc1[r] = ctxW2[(size_t)(m1 & (B_ - 1)) * V_ + nBase + l15];
  }
  wmma_tile2_ldsB(dec + (size_t)mT0 * 16 * H_, dec + (size_t)(mT0 + 1) * 16 * H_, H_,
                  ldsB, acc0, acc1);
  store_tile(out + (size_t)mT0 * 16 * V_ + nBase, V_, acc0);
  store_tile(out + (size_t)(mT0 + 1) * 16 * V_ + nBase, V_, acc1);
}

__global__ void __launch_bounds__(256) k_copy(const float* __restrict__ src,
                                              float* __restrict__ dst, int n) {
  int i = blockIdx.x * 256 + threadIdx.x;
  if (i < n) dst[i] = src[i];
}

// ---------------------------------------------------------------------------
extern "C" void kernel_launch(void* const* d_in, const int* in_sizes, int n_in,
                              void* d_out, int out_size, void* d_ws, size_t ws_size,
                              hipStream_t stream) {
  (void)in_sizes; (void)n_in; (void)out_size; (void)ws_size;

  const int*   tgt    = (const int*)d_in[0];
  const float* hidden = (const float*)d_in[1];
  const float* cell   = (const float*)d_in[2];
  const float* enc    = (const float*)d_in[3];
  const int*   smask  = (const int*)d_in[4];
  const float* emb    = (const float*)d_in[5];
  const float* Wih[2] = {(const float*)d_in[6],  (const float*)d_in[10]};
  const float* Whh[2] = {(const float*)d_in[7],  (const float*)d_in[11]};
  const float* bih[2] = {(const float*)d_in[8],  (const float*)d_in[12]};
  const float* bhh[2] = {(const float*)d_in[9],  (const float*)d_in[13]};
  /* d_in[14] == W1: provably unused (softmax weights independent of att) */
  const float* W2     = (const float*)d_in[15];
  float* out = (float*)d_out;

  // Workspace carve-up (floats): ~33.7 MB total
  float* ws    = (float*)d_ws;
  float* x0    = ws;                       // TB*H     embedding output
  float* ys0   = x0 + (size_t)TB * H_;     // TB*H     layer-0 hidden sequence
  float* dec   = ys0 + (size_t)TB * H_;    // TB*H     layer-1 hidden sequence
  float* G     = dec + (size_t)TB * H_;    // TB*4H    precomputed gate inputs
  float* hA    = G + (size_t)TB * G4H;     // B*H      h ping
  float* hB    = hA + B_ * H_;             // B*H      h pong
  float* cbuf  = hB + B_ * H_;             // B*H      c (in place, col-partitioned)
  float* ctx   = cbuf + B_ * H_;           // B*H
  float* ctxW2 = ctx + B_ * H_;            // B*V

  // 1. Embedding gather
  k_embed<<<(TB * H_) / 256, 256, 0, stream>>>(tgt, emb, x0);

  const float* xin[2] = {x0, ys0};
  float*       yout[2] = {ys0, dec};
  const size_t hsOff = (size_t)TB * V_;          // start of hs in d_out
  const size_t csOff = hsOff + 2 * B_ * H_;      // start of cs in d_out

  for (int l = 0; l < 2; ++l) {
    // 2. G = x @ W_ih^T + b_ih + b_hh   (WMMA, batched over all timesteps)
    k_gemm_gates<<<dim3(G4H / 16, TB / 256), 256, 0, stream>>>(
        xin[l], Wih[l], bih[l], bhh[l], G);
    // init recurrent state
    k_copy<<<(B_ * H_ + 255) / 256, 256, 0, stream>>>(hidden + (size_t)l * B_ * H_, hA, B_ * H_);
    k_copy<<<(B_ * H_ + 255) / 256, 256, 0, stream>>>(cell + (size_t)l * B_ * H_, cbuf, B_ * H_);
    // 3. sequential scan (h ping-pongs; T even so final h lands back in hA)
    for (int t = 0; t < T_; ++t) {
      float* hi = (t & 1) ? hB : hA;
      float* ho = (t & 1) ? hA : hB;
      k_lstm_step<<<32, 256, 0, stream>>>(G + (size_t)t * B_ * G4H, Whh[l], hi, ho,
                                          cbuf, yout[l] + (size_t)t * B_ * H_);
    }
    // final states -> output tail (hs then cs, stacked [L,B,H])
    k_copy<<<(B_ * H_ + 255) / 256, 256, 0, stream>>>(hA, out + hsOff + (size_t)l * B_ * H_, B_ * H_);
    k_copy<<<(B_ * H_ + 255) / 256, 256, 0, stream>>>(cbuf, out + csOff + (size_t)l * B_ * H_, B_ * H_);
  }

  // 4. Attention context (uniform weights over unmasked s; W1 drops out)
  k_ctx<<<(B_ * H_) / 256, 256, 0, stream>>>(enc, smask, ctx);

  // 5. t-independent half of vocab projection
  k_gemm_ctx<<<dim3(V_ / 16, 1), 256, 0, stream>>>(ctx, W2, ctxW2);

  // 6. Dominant GEMM: out = dec @ W2[:, :H]^T + ctxW2[b, :]
  k_gemm_out<<<dim3(V_ / 16, TB / 256), 256, 0, stream>>>(dec, W2, ctxW2, out);
}